// CorresAttention_66554813219085
// MI455X (gfx1250) — compile-verified
//
#include <hip/hip_runtime.h>
#include <math.h>

typedef __attribute__((ext_vector_type(2))) float v2f;
typedef __attribute__((ext_vector_type(8))) float v8f;
typedef __attribute__((ext_vector_type(4))) unsigned int v4u;
typedef __attribute__((ext_vector_type(8))) int v8i_;
typedef __attribute__((ext_vector_type(4))) int v4i_;

#define B_ 32
#define N_ 512
#define M_ 8192
#define C_ 3
#define K_ 8
#define CH_ 128          // x rows staged per TDM chunk
#define CHE_ (CH_ * 3)   // elements per chunk (384 f32 = 1536 B)

// ---------------------------------------------------------------------------
// TDM: 1-D tile load Global -> LDS via Tensor Data Mover (D# per ISA ch.8).
// Group0: count=1 | lds_addr | global_addr[56:0] | type=2
// Group1: data_size=4B, tensor_dim0 = tile_dim0 = nelems (1-D), rest 0.
// amdgpu-toolchain (clang-23) 6-arg builtin form:
//   (uint32x4 g0, int32x8 g1, int32x4 g2, int32x4 g3, int32x8 g4, i32 cpol)
// ---------------------------------------------------------------------------
__device__ __forceinline__ void tdm_load_1d(unsigned int lds_off,
                                            const void* gptr,
                                            unsigned int nelems)
{
    unsigned long long ga = (unsigned long long)(uintptr_t)gptr;
    v4u g0;
    g0.x = 1u;                                              // count=1 (valid)
    g0.y = lds_off;                                         // lds_addr (bytes)
    g0.z = (unsigned int)(ga & 0xFFFFFFFFull);              // global_addr lo
    g0.w = (unsigned int)((ga >> 32) & 0x01FFFFFFull)       // global_addr hi
         | 0x80000000u;                                     // type=2 ("image")
    v8i_ g1;
    g1[0] = (int)(2u << 16);                                // data_size=4B, wg_mask=0
    g1[1] = (int)((nelems & 0xFFFFu) << 16);                // tensor_dim0[15:0]
    g1[2] = (int)(((nelems >> 16) & 0xFFFFu) | (1u << 16)); // td0[31:16] | tensor_dim1=1
    g1[3] = (int)((nelems & 0xFFFFu) << 16);                // tile_dim0 (1-D tile)
    g1[4] = 0;                                              // tile_dim1=0, tile_dim2=0
    g1[5] = (int)nelems;                                    // tensor_dim0_stride lo
    g1[6] = 0;
    g1[7] = 0;
    v4i_ z4 = {0, 0, 0, 0};
    v8i_ z8 = {0, 0, 0, 0, 0, 0, 0, 0};
    __builtin_amdgcn_tensor_load_to_lds(g0, g1, z4, z4, z8, 0);
}

// ---------------------------------------------------------------------------
// Kernel 1: for each (batch, 64-row band of u), 4 waves x one 16-row tile:
//   - wave 0 TDM-streams x in 128-row chunks into double-buffered LDS
//     (overlapped with compute, S_WAIT_TENSORCNT + block barrier handoff)
//   - per 32-col step: 2x WMMA f32 16x16x4 -> dot(u,x) tiles (C=3 padded to 4)
//   - d = |u|^2 + |x|^2 - 2*dot -> LDS transpose -> per-row running top-8
//   - gather feat, softmax over K per channel, sum -> x_f (b,n,3)
// EXEC is all-ones at every WMMA (divergent code is outside the WMMA path).
// ---------------------------------------------------------------------------
__global__ __launch_bounds__(128) void knn_softsum_kernel(
    const float* __restrict__ u, const float* __restrict__ x,
    float* __restrict__ xf)
{
    __shared__ float sX[2][CHE_];        // double-buffered x chunk (TDM dest)
    __shared__ float sUn2[4][16];
    __shared__ float sTile[4][16][33];

    const int tid  = threadIdx.x;
    const int lane = tid & 31;
    const int wid  = tid >> 5;
    const int b    = blockIdx.x >> 3;                 // 8 bands per batch
    const int n0   = ((blockIdx.x & 7) << 6) + (wid << 4);

    const float* ub = u + (size_t)b * N_ * C_;
    const float* xb = x + (size_t)b * M_ * C_;

    const int  row = lane & 15;
    const bool hi  = lane >= 16;

    // A-matrix 16x4 f32: VGPR0 = {K0 | K2}, VGPR1 = {K1 | K3} across lane halves
    float u0 = ub[(n0 + row) * C_ + 0];
    float u1 = ub[(n0 + row) * C_ + 1];
    float u2 = ub[(n0 + row) * C_ + 2];
    if (!hi) sUn2[wid][lane] = u0 * u0 + u1 * u1 + u2 * u2;
    v2f a;
    a.x = hi ? u2 : u0;
    a.y = hi ? 0.0f : u1;

    // Prologue: DMA chunk 0 into buffer 0
    if (wid == 0) {
        tdm_load_1d((unsigned int)(uintptr_t)&sX[0][0], xb, CHE_);
        __builtin_amdgcn_s_wait_tensorcnt(0);
    }
    asm volatile("" ::: "memory");
    __syncthreads();

    float un2r[8];
    #pragma unroll
    for (int r = 0; r < 8; ++r) un2r[r] = sUn2[wid][r + (hi ? 8 : 0)];

    float bd[K_]; int bi[K_];
    #pragma unroll
    for (int j = 0; j < K_; ++j) { bd[j] = 3.4e38f; bi[j] = 0; }

    const int NCHUNK = M_ / CH_;   // 64
    for (int ch = 0; ch < NCHUNK; ++ch) {
        const int p = ch & 1;
        // Prefetch next chunk via TDM while computing on current one
        if (wid == 0 && (ch + 1) < NCHUNK)
            tdm_load_1d((unsigned int)(uintptr_t)&sX[p ^ 1][0],
                        xb + (size_t)(ch + 1) * CHE_, CHE_);

        const float* xs = &sX[p][0];

        #pragma unroll
        for (int st = 0; st < CH_ / 32; ++st) {      // 4 steps of 32 columns
            #pragma unroll
            for (int h2 = 0; h2 < 2; ++h2) {         // 2 WMMAs per step
                const int col = (st << 5) + (h2 << 4) + row;
                float x0 = xs[col * 3 + 0];
                float x1 = xs[col * 3 + 1];
                float x2 = xs[col * 3 + 2];
                float xn2 = x0 * x0 + x1 * x1 + x2 * x2;

                // B-matrix 4x16 f32: same lane-half K split as A
                v2f bm;
                bm.x = hi ? x2 : x0;
                bm.y = hi ? 0.0f : x1;

                v8f c = {};
                c = __builtin_amdgcn_wmma_f32_16x16x4_f32(
                        false, a, false, bm, (short)0, c, false, false);

                // d = |u|^2 + |x|^2 - 2*dot -> LDS (row-owner transpose)
                #pragma unroll
                for (int r = 0; r < 8; ++r)
                    sTile[wid][r + (hi ? 8 : 0)][(h2 << 4) + row] =
                        un2r[r] + xn2 - 2.0f * c[r];
            }
            // same-wave LDS producer/consumer: DS is in-order per wave;
            // wave_barrier stops compiler reordering only.
            __builtin_amdgcn_wave_barrier();
            if (lane < 16) {
                const int base = (ch << 7) + (st << 5);   // global m of col 0
                #pragma unroll 1
                for (int j = 0; j < 32; ++j) {
                    float d = sTile[wid][lane][j];
                    if (d < bd[K_ - 1]) {
                        bd[K_ - 1] = d; bi[K_ - 1] = base + j;
                        #pragma unroll
                        for (int t = K_ - 1; t > 0; --t) {
                            if (bd[t] < bd[t - 1]) {
                                float td = bd[t]; bd[t] = bd[t - 1]; bd[t - 1] = td;
                                int   ti = bi[t]; bi[t] = bi[t - 1]; bi[t - 1] = ti;
                            }
                        }
                    }
                }
            }
            __builtin_amdgcn_wave_barrier();
        }

        // double-buffer handoff: next iteration's DMA target is the buffer
        // all waves just finished reading
        if (wid == 0) __builtin_amdgcn_s_wait_tensorcnt(0);
        asm volatile("" ::: "memory");
        __syncthreads();
    }

    if (lane < 16) {
        // gather neighbors, softmax over K per channel, sum over K
        float f[K_][C_];
        #pragma unroll
        for (int k = 0; k < K_; ++k) {
            const float* xr = xb + (size_t)bi[k] * C_;
            f[k][0] = xr[0]; f[k][1] = xr[1]; f[k][2] = xr[2];
        }
        float* o = xf + ((size_t)b * N_ + n0 + lane) * C_;
        #pragma unroll
        for (int cc = 0; cc < C_; ++cc) {
            float mx = f[0][cc];
            #pragma unroll
            for (int k = 1; k < K_; ++k) mx = fmaxf(mx, f[k][cc]);
            float e[K_]; float s = 0.0f;
            #pragma unroll
            for (int k = 0; k < K_; ++k) { e[k] = __expf(f[k][cc] - mx); s += e[k]; }
            float r = 0.0f;
            #pragma unroll
            for (int k = 0; k < K_; ++k) r += e[k] / s;
            o[cc] = r;
        }
    }
}

// ---------------------------------------------------------------------------
// Kernel 2: q/k/v projection + online-softmax attention + out-proj + conv1.
// One block per (batch, half of N). k/v for the whole batch staged in LDS.
// Emits h[b,n] and per-block partial (sum, sumsq) for the LayerNorm stats.
// ---------------------------------------------------------------------------
__global__ __launch_bounds__(256) void attn_kernel(
    const float* __restrict__ u,  const float* __restrict__ xf,
    const float* __restrict__ ipw, const float* __restrict__ ipb,
    const float* __restrict__ ow,  const float* __restrict__ ob,
    const float* __restrict__ c1w,
    float* __restrict__ h, float* __restrict__ part)
{
    __shared__ float sK[N_][3];
    __shared__ float sV[N_][3];
    __shared__ float red[256];

    const int tid = threadIdx.x;
    const int b   = blockIdx.x >> 1;
    const int n   = ((blockIdx.x & 1) << 8) + tid;

    for (int m = tid; m < N_; m += 256) {
        const float* xr = xf + ((size_t)b * N_ + m) * C_;
        float x0 = xr[0], x1 = xr[1], x2 = xr[2];
        #pragma unroll
        for (int c = 0; c < 3; ++c) {
            const float* wk = ipw + (size_t)(3 + c) * C_;
            const float* wv = ipw + (size_t)(6 + c) * C_;
            sK[m][c] = x0 * wk[0] + x1 * wk[1] + x2 * wk[2] + ipb[3 + c];
            sV[m][c] = x0 * wv[0] + x1 * wv[1] + x2 * wv[2] + ipb[6 + c];
        }
    }
    __syncthreads();

    const float* ur = u + ((size_t)b * N_ + n) * C_;
    float t0 = ur[0], t1 = ur[1], t2 = ur[2];
    float q[3];
    #pragma unroll
    for (int c = 0; c < 3; ++c) {
        const float* wq = ipw + (size_t)c * C_;
        q[c] = t0 * wq[0] + t1 * wq[1] + t2 * wq[2] + ipb[c];
    }

    const float rs3 = 0.57735026918962576f; // 1/sqrt(3)
    float mx = -3.4e38f, den = 0.0f, a0 = 0.0f, a1 = 0.0f, a2 = 0.0f;
    for (int m = 0; m < N_; ++m) {
        float s = (q[0] * sK[m][0] + q[1] * sK[m][1] + q[2] * sK[m][2]) * rs3;
        if (s > mx) {
            float r = __expf(mx - s);
            den *= r; a0 *= r; a1 *= r; a2 *= r; mx = s;
        }
        float e = __expf(s - mx);
        den += e;
        a0 += e * sV[m][0]; a1 += e * sV[m][1]; a2 += e * sV[m][2];
    }
    float uf0 = a0 / den, uf1 = a1 / den, uf2 = a2 / den;

    float y[3];
    #pragma unroll
    for (int c = 0; c < 3; ++c)
        y[c] = ow[c * 3 + 0] * uf0 + ow[c * 3 + 1] * uf1 + ow[c * 3 + 2] * uf2 + ob[c];

    float hv = c1w[0] * y[0] + c1w[1] * y[1] + c1w[2] * y[2];
    h[(size_t)b * N_ + n] = hv;

    red[tid] = hv;
    __syncthreads();
    for (int s = 128; s > 0; s >>= 1) {
        if (tid < s) red[tid] += red[tid + s];
        __syncthreads();
    }
    float bsum = red[0];
    __syncthreads();
    red[tid] = hv * hv;
    __syncthreads();
    for (int s = 128; s > 0; s >>= 1) {
        if (tid < s) red[tid] += red[tid + s];
        __syncthreads();
    }
    if (tid == 0) {
        part[blockIdx.x * 2 + 0] = bsum;
        part[blockIdx.x * 2 + 1] = red[0];
    }
}

// ---------------------------------------------------------------------------
// Kernel 3: LayerNorm (per batch over N) + exact gelu + conv2 + sigmoid.
// ---------------------------------------------------------------------------
__global__ __launch_bounds__(256) void final_kernel(
    const float* __restrict__ h, const float* __restrict__ part,
    const float* __restrict__ lnw, const float* __restrict__ lnb,
    const float* __restrict__ c2w, const float* __restrict__ c2b,
    float* __restrict__ out)
{
    const int b = blockIdx.x >> 1;
    const int n = ((blockIdx.x & 1) << 8) + threadIdx.x;

    float sum = part[(2 * b + 0) * 2 + 0] + part[(2 * b + 1) * 2 + 0];
    float sq  = part[(2 * b + 0) * 2 + 1] + part[(2 * b + 1) * 2 + 1];
    float mean = sum * (1.0f / (float)N_);
    float var  = sq * (1.0f / (float)N_) - mean * mean;

    float hv = h[(size_t)b * N_ + n];
    float hn = (hv - mean) * rsqrtf(var + 1e-5f) * lnw[n] + lnb[n];
    float g  = 0.5f * hn * (1.0f + erff(hn * 0.70710678118654752f)); // exact gelu
    float z  = g * c2w[0] + c2b[0];
    out[(size_t)b * N_ + n] = 1.0f / (1.0f + __expf(-z));
}

// ---------------------------------------------------------------------------
extern "C" void kernel_launch(void* const* d_in, const int* in_sizes, int n_in,
                              void* d_out, int out_size, void* d_ws, size_t ws_size,
                              hipStream_t stream)
{
    const float* u   = (const float*)d_in[0];   // (32,512,3)
    const float* x   = (const float*)d_in[1];   // (32,8192,3)
    const float* ipw = (const float*)d_in[2];   // (9,3)
    const float* ipb = (const float*)d_in[3];   // (9,)
    const float* ow  = (const float*)d_in[4];   // (3,3)
    const float* ob  = (const float*)d_in[5];   // (3,)
    const float* c1w = (const float*)d_in[6];   // (1,3)
    const float* lnw = (const float*)d_in[7];   // (1,512)
    const float* lnb = (const float*)d_in[8];   // (1,512)
    const float* c2w = (const float*)d_in[9];   // (1,1)
    const float* c2b = (const float*)d_in[10];  // (1,)
    float* out = (float*)d_out;                 // (32,512)

    float* ws   = (float*)d_ws;
    float* xfb  = ws;                            // 32*512*3 = 49152 floats
    float* hbuf = xfb + (size_t)B_ * N_ * C_;    // 16384 floats
    float* part = hbuf + (size_t)B_ * N_;        // 128 floats

    // Stage 1: TDM-streamed WMMA distance tiles + top-8 + softmax-sum -> x_f
    knn_softsum_kernel<<<dim3(B_ * (N_ / 64)), dim3(128), 0, stream>>>(u, x, xfb);

    // Stage 2: attention collapse -> h + LN partials
    attn_kernel<<<dim3(B_ * 2), dim3(256), 0, stream>>>(
        u, xfb, ipw, ipb, ow, ob, c1w, hbuf, part);

    // Stage 3: LN + gelu + conv2 + sigmoid -> out
    final_kernel<<<dim3(B_ * 2), dim3(256), 0, stream>>>(
        hbuf, part, lnw, lnb, c2w, c2b, out);
}